// GraphAttentionLayer_45251775430863
// MI455X (gfx1250) — compile-verified
//
#include <hip/hip_runtime.h>
#include <math.h>

// Problem constants from the reference
static constexpr int   B  = 42;
static constexpr int   N  = 77;
static constexpr int   FP = 32;
static constexpr int   NF = 12;
static constexpr float NEGC  = -9000000000000000.0f;
static constexpr float LKY   = 0.2f;

typedef float v2f __attribute__((ext_vector_type(2)));
typedef float v8f __attribute__((ext_vector_type(8)));

// ---------------------------------------------------------------------------
// Kernel 1 (one wave32): compute the reduction scalars and C = WQ @ WK^T / sqrt(40)
// via V_WMMA_F32_16X16X4_F32 (M,N padded 2->16; K=40 done as 10 chained x4 steps).
// ISA layouts (CDNA5 ISA §7.12.2):
//   A 16x4 f32 : lanes 0-15 row M=lane, v0=K0,v1=K1 ; lanes16-31 row M=lane-16, v0=K2,v1=K3
//   B 4x16 f32 : lanes 0-15 col N=lane, v0=row K0,v1=row K1 ; lanes16-31 v0=K2,v1=K3
//   D 16x16 f32: acc[r] = D[M=r][N=lane] (lanes 0-15)
// Workspace layout (floats): [0]=c1 [1]=c2 [2]=c1s [3]=c2s [4]=C00 [5]=C01 [6]=C10 [7]=C11
// ---------------------------------------------------------------------------
__global__ __launch_bounds__(32)
void gat_setup_kernel(const float* __restrict__ W,  const float* __restrict__ WS,
                      const float* __restrict__ a,  const float* __restrict__ aS,
                      const float* __restrict__ WQ, const float* __restrict__ WK,
                      float* __restrict__ ws) {
  const int lane = threadIdx.x;
  const int half = lane >> 4;      // 0: K pair {0,1}, 1: K pair {2,3}
  const int m    = lane & 15;      // padded row / col index

  // --- WMMA accumulation for C (EXEC stays all-ones: branch-free selects only)
  v8f acc = {0.f, 0.f, 0.f, 0.f, 0.f, 0.f, 0.f, 0.f};
  #pragma unroll
  for (int kk = 0; kk < 40; kk += 4) {
    const int k0 = kk + half * 2;
    v2f av, bv;
    av.x = (m < 2) ? WQ[m * 40 + k0]     : 0.0f;   // A = WQ padded to 16x4 slice
    av.y = (m < 2) ? WQ[m * 40 + k0 + 1] : 0.0f;
    bv.x = (m < 2) ? WK[m * 40 + k0]     : 0.0f;   // B = WK^T padded (B[k][n] = WK[n][k])
    bv.y = (m < 2) ? WK[m * 40 + k0 + 1] : 0.0f;
    acc = __builtin_amdgcn_wmma_f32_16x16x4_f32(false, av, false, bv,
                                                (short)0, acc, false, false);
  }

  // --- scalar dot products (Fp == 32 == wave width)
  float v0 = W[lane]  * a[lane];
  float v1 = W[lane]  * a[32 + lane];
  float v2 = WS[lane] * aS[lane];
  float v3 = WS[lane] * aS[32 + lane];
  #pragma unroll
  for (int off = 16; off > 0; off >>= 1) {
    v0 += __shfl_xor(v0, off, 32);
    v1 += __shfl_xor(v1, off, 32);
    v2 += __shfl_xor(v2, off, 32);
    v3 += __shfl_xor(v3, off, 32);
  }

  const float scale = 0.15811388300841897f;  // 1/sqrt(40)
  if (lane == 0) {
    ws[0] = v0; ws[1] = v1; ws[2] = v2; ws[3] = v3;
    ws[4] = acc[0] * scale;   // C00 = D[0][0]
    ws[6] = acc[1] * scale;   // C10 = D[1][0]
  }
  if (lane == 1) {
    ws[5] = acc[0] * scale;   // C01 = D[0][1]
    ws[7] = acc[1] * scale;   // C11 = D[1][1]
  }
}

// ---------------------------------------------------------------------------
// Kernel 2: one wave32 per (b,i) row. Each lane owns j in {lane, lane+32, lane+64}.
//  Pass 1: masked GAT score row -> wave softmax over j (max + sum via shfl_xor).
//  Pass 2: per (j): 12-wide masked softmax over f dotted with ext_i -> u accumulation;
//          plus t = sum_j att*x_j. Lane f writes elu(t*W[f]) and elu(u*WV[f]).
// ---------------------------------------------------------------------------
__global__ __launch_bounds__(256)
void gat_row_kernel(const float* __restrict__ x,    const int* __restrict__ adj,
                    const float* __restrict__ ext,  const float* __restrict__ side,
                    const float* __restrict__ W,    const float* __restrict__ WV,
                    const float* __restrict__ ws,   float* __restrict__ out) {
  const int wid  = (int)((blockIdx.x * blockDim.x + threadIdx.x) >> 5);
  const int lane = threadIdx.x & 31;
  if (wid >= B * N) return;
  const int b = wid / N;
  const int i = wid - b * N;

  const float c1  = ws[0], c2  = ws[1], c1s = ws[2], c2s = ws[3];
  const float C00 = ws[4], C01 = ws[5], C10 = ws[6], C11 = ws[7];

  const float* __restrict__ xb   = x    + b * N;
  const float* __restrict__ sb   = side + b * N;
  const int*   __restrict__ arow = adj  + i * N;
  const float* __restrict__ extb = ext  + (size_t)b * N * NF;

  const float xi   = xb[i];
  const float s1i  = xi * c1;
  const float s1si = sb[i] * c1s;

  // ext_i row (12 f32 = 3x b128, wave-uniform address)
  float ei[NF];
  {
    const float4* ep = (const float4*)(extb + i * NF);
    float4 e0 = ep[0], e1 = ep[1], e2 = ep[2];
    ei[0]=e0.x; ei[1]=e0.y; ei[2]=e0.z; ei[3]=e0.w;
    ei[4]=e1.x; ei[5]=e1.y; ei[6]=e1.z; ei[7]=e1.w;
    ei[8]=e2.x; ei[9]=e2.y; ei[10]=e2.z; ei[11]=e2.w;
  }
  const float xiC01 = xi * C01;
  float Pf[NF], Rf[NF];
  #pragma unroll
  for (int f = 0; f < NF; ++f) { Pf[f] = xi * C00 * ei[f]; Rf[f] = C10 * ei[f]; }

  // ---- pass 1: GAT row scores for this wave's j's
  float apre[3], xjv[3];
  float mx = NEGC;
  #pragma unroll
  for (int k = 0; k < 3; ++k) {
    const int j = lane + k * 32;
    float ap = NEGC, xj = 0.0f;
    if (j < N) {
      xj = xb[j];
      float e  = s1i  + xj * c2;      e  = (e  >= 0.0f) ? e  : LKY * e;   // leaky_relu
      float es = s1si + sb[j] * c2s;  es = (es >= 0.0f) ? es : LKY * es;
      const float v = (arow[j] > 0) ? (e + es) : NEGC;  // mask; (NEG+NEG)>0 is false too
      ap = (v > 0.0f) ? v : NEGC;
      mx = fmaxf(mx, ap);
    }
    apre[k] = ap; xjv[k] = xj;
  }
  #pragma unroll
  for (int off = 16; off > 0; off >>= 1) mx = fmaxf(mx, __shfl_xor(mx, off, 32));

  float pv[3];
  float psum = 0.0f;
  #pragma unroll
  for (int k = 0; k < 3; ++k) {
    const int j = lane + k * 32;
    pv[k] = (j < N) ? expf(apre[k] - mx) : 0.0f;
    psum += pv[k];
  }
  #pragma unroll
  for (int off = 16; off > 0; off >>= 1) psum += __shfl_xor(psum, off, 32);
  const float attScale = 1.0f / psum;

  // ---- pass 2: per-j 12-wide f-softmax dotted with ext_i
  float accT = 0.0f, accU = 0.0f;
  #pragma unroll
  for (int k = 0; k < 3; ++k) {
    const int j = lane + k * 32;
    if (j < N) {
      const float att = pv[k] * attScale;
      const float xj  = xjv[k];
      accT += att * xj;

      float ej[NF];
      {
        const float4* ep = (const float4*)(extb + j * NF);
        float4 a0 = ep[0], a1 = ep[1], a2 = ep[2];
        ej[0]=a0.x; ej[1]=a0.y; ej[2]=a0.z; ej[3]=a0.w;
        ej[4]=a1.x; ej[5]=a1.y; ej[6]=a1.z; ej[7]=a1.w;
        ej[8]=a2.x; ej[9]=a2.y; ej[10]=a2.z; ej[11]=a2.w;
      }
      float dv[NF];
      float dmax = NEGC;
      #pragma unroll
      for (int f = 0; f < NF; ++f) {
        float d = Pf[f] + xiC01 * ej[f] + xj * (Rf[f] + C11 * ej[f]);
        d = (d > 0.0f) ? d : NEGC;
        dv[f] = d;
        dmax = fmaxf(dmax, d);
      }
      float wsum = 0.0f, wdot = 0.0f;
      #pragma unroll
      for (int f = 0; f < NF; ++f) {
        const float w = expf(dv[f] - dmax);
        wsum += w;
        wdot += w * ei[f];
      }
      accU += att * (wdot / wsum);
    }
  }
  #pragma unroll
  for (int off = 16; off > 0; off >>= 1) {
    accT += __shfl_xor(accT, off, 32);
    accU += __shfl_xor(accU, off, 32);
  }

  // ---- outputs: lane == feature index f (FP == 32)
  float o1 = accT * W[lane];
  o1 = (o1 > 0.0f) ? o1 : expm1f(o1);              // elu, alpha=1
  float o2 = accU * WV[lane];
  o2 = (o2 > 0.0f) ? o2 : expm1f(o2);
  const size_t base = (size_t)wid * FP;
  out[base + lane] = o1;
  out[(size_t)B * N * FP + base + lane] = o2;
}

// ---------------------------------------------------------------------------
extern "C" void kernel_launch(void* const* d_in, const int* in_sizes, int n_in,
                              void* d_out, int out_size, void* d_ws, size_t ws_size,
                              hipStream_t stream) {
  const float* x    = (const float*)d_in[0];
  const int*   adj  = (const int*)  d_in[1];
  const float* ext  = (const float*)d_in[2];
  const float* side = (const float*)d_in[3];
  const float* W    = (const float*)d_in[4];
  const float* WS   = (const float*)d_in[5];
  const float* a    = (const float*)d_in[6];
  const float* aS   = (const float*)d_in[7];
  const float* WQ   = (const float*)d_in[8];
  const float* WK   = (const float*)d_in[9];
  const float* WV   = (const float*)d_in[10];
  float* wsf = (float*)d_ws;
  float* out = (float*)d_out;

  gat_setup_kernel<<<1, 32, 0, stream>>>(W, WS, a, aS, WQ, WK, wsf);

  const int waves  = B * N;                  // 3234 rows, one wave32 each
  const int blocks = (waves + 7) / 8;        // 8 waves per 256-thread block
  gat_row_kernel<<<blocks, 256, 0, stream>>>(x, adj, ext, side, W, WV, wsf, out);
}